// QueryAndGroup_40261023433415
// MI455X (gfx1250) — compile-verified
//
#include <hip/hip_runtime.h>

typedef float v2f __attribute__((ext_vector_type(2)));
typedef float v8f __attribute__((ext_vector_type(8)));

#define NB        4
#define NPTS      16384
#define NS        2048
#define NC        64
#define HALF_N    8192
#define NTILE     (HALF_N / 16)
#define BQ_WAVES  4
#define BQ_THREADS (BQ_WAVES * 32)
#define WGS_PER_BATCH 32
#define R2      (0.1f * 0.1f)
#define R2_PAD  (0.1f * 0.1f + 1.0e-5f)   // guard band for expanded-form rounding
#define GG_WAVES 8

__device__ __forceinline__ float rlane_f(float x, int l) {
  return __int_as_float(__builtin_amdgcn_readlane(__float_as_int(x), l));
}

// Cold path: exact recheck + in-order append of candidate bits for one query.
// noinline -> exactly one copy in the binary (32 call sites otherwise explode
// the I-footprint). Count is passed by value / returned so the caller's cnt[]
// array never has its address taken and stays in registers.
__device__ __attribute__((noinline)) int append_hits(
    unsigned bits, int n0_local, int n0_global, int qv,
    float qxl, float qyl, float qzl,          // per-lane query coords
    const float4* pts, int* buf, int c, int lane)
{
  const float qx = rlane_f(qxl, qv);          // qv is uniform: v_readlane ok
  const float qy = rlane_f(qyl, qv);
  const float qz = rlane_f(qzl, qv);
  while (bits && c < 32) {
    const int t = __builtin_ctz(bits);
    bits &= bits - 1u;
    const float4 P = pts[n0_local + t];
    const float dx = qx - P.x, dy = qy - P.y, dz = qz - P.z;
    const float d2 = dx * dx + dy * dy + dz * dz;   // exact form, matches reference
    if (d2 < R2) {
      if (lane == 0) buf[c] = n0_global + t;        // single-lane LDS write
      ++c;
    }
  }
  return c;
}

__global__ __launch_bounds__(BQ_THREADS)
void ball_query_wmma_kernel(const float* __restrict__ xyz,
                            const float* __restrict__ new_xyz,
                            int* __restrict__ ball_idx)
{
  __shared__ float4 s_pts[HALF_N];               // 128 KB: (x, y, z, |p|^2)
  __shared__ int    s_buf[BQ_WAVES][16][32];     // 8 KB per-wave hit buffers

  const int tid  = threadIdx.x;
  const int lane = tid & 31;
  const int w    = tid >> 5;
  const int b    = blockIdx.x / WGS_PER_BATCH;
  const int part = blockIdx.x % WGS_PER_BATCH;
  const int s0   = (part * BQ_WAVES + w) * 16;   // this wave owns queries s0..s0+15

  // Lanes l and l+16 both hold query s0 + (l & 15).
  const int   q  = s0 + (lane & 15);
  const float qx = new_xyz[(b * NS + q) * 3 + 0];
  const float qy = new_xyz[(b * NS + q) * 3 + 1];
  const float qz = new_xyz[(b * NS + q) * 3 + 2];
  const float q2 = qx * qx + qy * qy + qz * qz;

  // A (16x4): row m = (-2x, -2y, -2z, 1).  Lanes 0-15: K=0,1;  lanes 16-31: K=2,3.
  v2f amat;
  amat.x = (lane < 16) ? (-2.0f * qx) : (-2.0f * qz);
  amat.y = (lane < 16) ? (-2.0f * qy) : 1.0f;

  // C (16x16 f32): VGPR v holds |q|^2 of query (lane<16 ? v : v+8), constant over N.
  v8f cmat;
#pragma unroll
  for (int v = 0; v < 8; ++v) {
    const int sel = ((lane < 16) ? v : (v + 8)) << 2;
    cmat[v] = __int_as_float(__builtin_amdgcn_ds_bpermute(sel, __float_as_int(q2)));
  }

  int cnt[16];
#pragma unroll
  for (int i = 0; i < 16; ++i) cnt[i] = 0;
  unsigned doneMask = 0;

  const int boff = (lane < 16) ? 0 : 8;          // byte offset within float4

  for (int pass = 0; pass < 2; ++pass) {
    __syncthreads();
    // Cooperative LDS stage of points [pass*HALF_N, pass*HALF_N + HALF_N)
    for (int i = tid; i < HALF_N; i += BQ_THREADS) {
      const int g = pass * HALF_N + i;
      __builtin_prefetch(&xyz[(b * NPTS + g + BQ_THREADS) * 3], 0, 1);
      const float px = xyz[(b * NPTS + g) * 3 + 0];
      const float py = xyz[(b * NPTS + g) * 3 + 1];
      const float pz = xyz[(b * NPTS + g) * 3 + 2];
      s_pts[i] = make_float4(px, py, pz, px * px + py * py + pz * pz);
    }
    __syncthreads();

    if (doneMask == 0xffffu) continue;   // still hit the barriers above each pass

    // B-operand fetch for one 16-point tile:
    //   VGPR0 lanes0-15 = x (K=0), lanes16-31 = z (K=2)
    //   VGPR1 lanes0-15 = y (K=1), lanes16-31 = |p|^2 (K=3)
    // AoS float4 layout -> conflict-free; two tiles merge into ds_load_2addr_b64.
    auto loadB = [&](int t) -> v2f {
      const int nl = t * 16 + (lane & 15);
      const float2 bv = *(const float2*)((const char*)&s_pts[nl] + boff);
      v2f r; r.x = bv.x; r.y = bv.y; return r;
    };

    // Software pipeline: 2 tiles (32 points) per iteration, B double-buffered so
    // the ds_load latency of tiles t+2/t+3 hides under the WMMAs/cmps of t/t+1.
    v2f bcur0 = loadB(0);
    v2f bcur1 = loadB(1);
    for (int t = 0; t < NTILE; t += 2) {
      const int tp = (t + 2 < NTILE) ? (t + 2) : (NTILE - 2);
      v2f bnxt0 = loadB(tp);
      v2f bnxt1 = loadB(tp + 1);

      // Two independent WMMAs (distinct B/D, shared A): no RAW hazard, back-to-back.
      v8f d0 = __builtin_amdgcn_wmma_f32_16x16x4_f32(
          false, amat, false, bcur0, (short)0, cmat, false, false);
      v8f d1 = __builtin_amdgcn_wmma_f32_16x16x4_f32(
          false, amat, false, bcur1, (short)0, cmat, false, false);

      unsigned m0[8], m1[8]; unsigned any = 0;
#pragma unroll
      for (int v = 0; v < 8; ++v) {
        m0[v] = (unsigned)__builtin_amdgcn_ballot_w32(d0[v] < R2_PAD);
        m1[v] = (unsigned)__builtin_amdgcn_ballot_w32(d1[v] < R2_PAD);
        any |= m0[v] | m1[v];
      }

      if (any) {
        // Tile t first, then tile t+1: preserves per-query ascending index order.
#pragma unroll
        for (int half = 0; half < 2; ++half) {
          const unsigned* m = half ? m1 : m0;
          const int n0l = (t + half) * 16;
          const int n0g = pass * HALF_N + n0l;
#pragma unroll
          for (int v = 0; v < 8; ++v) {
            const unsigned lo = m[v] & 0xffffu;      // query v, points n0..n0+15
            if (lo && cnt[v] < 32) {
              cnt[v] = append_hits(lo, n0l, n0g, v, qx, qy, qz,
                                   s_pts, &s_buf[w][v][0], cnt[v], lane);
              if (cnt[v] >= 32) doneMask |= 1u << v;
            }
            const unsigned hi = m[v] >> 16;          // query v+8, same points
            if (hi && cnt[v + 8] < 32) {
              cnt[v + 8] = append_hits(hi, n0l, n0g, v + 8, qx, qy, qz,
                                       s_pts, &s_buf[w][v + 8][0], cnt[v + 8], lane);
              if (cnt[v + 8] >= 32) doneMask |= 1u << (v + 8);
            }
          }
        }
        if (doneMask == 0xffffu) break;              // all 16 queries filled
      }

      bcur0 = bnxt0;
      bcur1 = bnxt1;
    }
  }

  // Finalize: pad with first hit (or 0 if none), write (B,S,32) int32 ball indices.
  const int outBase = (b * NS + s0) * 32;
#pragma unroll
  for (int qv = 0; qv < 16; ++qv) {
    const int c     = cnt[qv];
    const int first = (c > 0) ? s_buf[w][qv][0] : 0;
    const int mine  = s_buf[w][qv][lane];
    ball_idx[outBase + qv * 32 + lane] = (lane < c) ? mine : first;
  }
}

__global__ __launch_bounds__(GG_WAVES * 32)
void group_gather_kernel(const float* __restrict__ xyz,
                         const float* __restrict__ new_xyz,
                         const float* __restrict__ features,
                         const int* __restrict__ fps_idx,
                         const int* __restrict__ ball_idx,
                         float* __restrict__ out)
{
  const int lane = threadIdx.x & 31;
  const int w    = threadIdx.x >> 5;
  const int gs   = blockIdx.x * GG_WAVES + w;    // 0 .. NB*NS-1, one wave per query
  const int b    = gs / NS;
  const int s    = gs % NS;

  const float nx = new_xyz[(b * NS + s) * 3 + 0];
  const float ny = new_xyz[(b * NS + s) * 3 + 1];
  const float nz = new_xyz[(b * NS + s) * 3 + 2];

  const size_t chStride = (size_t)NS * 33;

  for (int k = lane; k < 33; k += 32) {          // lane 0 also covers k=32
    const int i = (k == 0) ? fps_idx[b * NS + s]
                           : ball_idx[(b * NS + s) * 32 + (k - 1)];
    const float px = xyz[(b * NPTS + i) * 3 + 0];
    const float py = xyz[(b * NPTS + i) * 3 + 1];
    const float pz = xyz[(b * NPTS + i) * 3 + 2];

    size_t base = (((size_t)b * 70) * NS + s) * 33 + (size_t)k;   // ch = 0
    out[base + 0 * chStride] = px;               // raw grouped xyz
    out[base + 1 * chStride] = py;
    out[base + 2 * chStride] = pz;
    out[base + 3 * chStride] = px - nx;          // centered grouped xyz
    out[base + 4 * chStride] = py - ny;
    out[base + 5 * chStride] = pz - nz;

    const float* f = features + ((size_t)b * NC) * NPTS + (size_t)i;
    size_t o = base + 6 * chStride;
#pragma unroll 8
    for (int c2 = 0; c2 < NC; ++c2) {            // grouped features
      out[o] = f[(size_t)c2 * NPTS];
      o += chStride;
    }
  }
}

extern "C" void kernel_launch(void* const* d_in, const int* in_sizes, int n_in,
                              void* d_out, int out_size, void* d_ws, size_t ws_size,
                              hipStream_t stream) {
  (void)in_sizes; (void)n_in; (void)out_size; (void)ws_size;
  const float* xyz      = (const float*)d_in[0];
  const float* new_xyz  = (const float*)d_in[1];
  const float* features = (const float*)d_in[2];
  const int*   fps_idx  = (const int*)d_in[3];
  int* ball = (int*)d_ws;                        // B*S*32 int32 = 1 MB scratch

  ball_query_wmma_kernel<<<NB * WGS_PER_BATCH, BQ_THREADS, 0, stream>>>(
      xyz, new_xyz, ball);
  group_gather_kernel<<<(NB * NS) / GG_WAVES, GG_WAVES * 32, 0, stream>>>(
      xyz, new_xyz, features, fps_idx, ball, (float*)d_out);
}